// MultiTokenConceptLayer_68083821576472
// MI455X (gfx1250) — compile-verified
//
#include <hip/hip_runtime.h>
#include <stdint.h>

typedef float v2f __attribute__((ext_vector_type(2)));
typedef float v8f __attribute__((ext_vector_type(8)));

#define B_DIM   16
#define S_DIM   4096
#define D_DIM   1024
#define K_KEYS  65536

#define LDS_STRIDE 20   // floats; 80B rows -> 16B-aligned for async b128 LDS writes

#if __has_builtin(__builtin_amdgcn_global_load_async_to_lds_b128)
#define HAVE_ASYNC_LDS 1
#else
#define HAVE_ASYNC_LDS 0
#endif

// Builtin signature (from hipcc diagnostic): the pointers are gcc-vector int4
// pointers in AS(1) (global, printed as __device__) and AS(3) (LDS).
typedef int v4i_vs __attribute__((vector_size(16)));
typedef __attribute__((address_space(1))) v4i_vs gv4_t;
typedef __attribute__((address_space(3))) v4i_vs lv4_t;

// ---------------------------------------------------------------------------
// CDNA5 async global->LDS b128 (ASYNCcnt path). Address-space casts go via
// integer round-trip (AS3 pointers are 32-bit; flat LDS addrs truncate to the
// LDS offset in the low 32 bits per the aperture rules).
// ---------------------------------------------------------------------------
__device__ __forceinline__ void async_copy_b128(const float* gsrc, float* ldst) {
#if HAVE_ASYNC_LDS
    gv4_t* g = reinterpret_cast<gv4_t*>(reinterpret_cast<uintptr_t>(gsrc));
    lv4_t* l = reinterpret_cast<lv4_t*>(
                   static_cast<uint32_t>(reinterpret_cast<uintptr_t>(ldst)));
    __builtin_amdgcn_global_load_async_to_lds_b128(g, l, /*offset=*/0, /*cpol=*/0);
#else
    float4 v = *(const float4*)gsrc;
    ldst[0] = v.x; ldst[1] = v.y; ldst[2] = v.z; ldst[3] = v.w;
#endif
}

__device__ __forceinline__ void wait_async_all() {
#if HAVE_ASYNC_LDS
#if __has_builtin(__builtin_amdgcn_s_wait_asynccnt)
    __builtin_amdgcn_s_wait_asynccnt(0);
#else
    asm volatile("s_wait_asynccnt 0x0" ::: "memory");
#endif
#endif
}

// ---------------------------------------------------------------------------
// Kernel 0: zero the 16 u64 argmax slots (deterministic init every call)
// ---------------------------------------------------------------------------
__global__ void init_best_kernel(unsigned long long* __restrict__ best) {
    if (threadIdx.x < B_DIM) best[threadIdx.x] = 0ull;
}

// ---------------------------------------------------------------------------
// Kernel 1: layer_out = hidden @ W^T + bias into the (B, S+1, D) output,
// rows [0, S). 128x128 tile / workgroup, 8 waves, WMMA f32 16x16x4,
// double-buffered async global->LDS staging.
// ---------------------------------------------------------------------------
__global__ __launch_bounds__(256)
void gemm_bias_kernel(const float* __restrict__ hidden,
                      const float* __restrict__ W,
                      const float* __restrict__ bias,
                      float* __restrict__ out)
{
    __shared__ float As[2][128][LDS_STRIDE];   // [buf][m_local][k]
    __shared__ float Bs[2][128][LDS_STRIDE];   // [buf][n_local][k] (W row-major (n,k))

    const int tid  = threadIdx.x;
    const int wave = tid >> 5;
    const int lane = tid & 31;
    const int half = lane >> 4;     // 0: lanes 0-15, 1: lanes 16-31
    const int idx  = lane & 15;

    const int blockM = blockIdx.x * 128;
    const int blockN = blockIdx.y * 128;
    const int batch  = blockM / S_DIM;         // uniform: 4096 % 128 == 0

    v8f acc[8] = {};

    const int ldrow = tid >> 2;                // 0..63
    const int ldcol = (tid & 3) * 4;           // 0,4,8,12

    const float* aptr0 = hidden + (size_t)(blockM + ldrow)      * D_DIM + ldcol;
    const float* aptr1 = hidden + (size_t)(blockM + ldrow + 64) * D_DIM + ldcol;
    const float* bptr0 = W      + (size_t)(blockN + ldrow)      * D_DIM + ldcol;
    const float* bptr1 = W      + (size_t)(blockN + ldrow + 64) * D_DIM + ldcol;

    // stage one 16-deep K slab into LDS buffer `buf`
    auto issue_stage = [&](int buf, int k0) {
        async_copy_b128(aptr0 + k0, &As[buf][ldrow     ][ldcol]);
        async_copy_b128(aptr1 + k0, &As[buf][ldrow + 64][ldcol]);
        async_copy_b128(bptr0 + k0, &Bs[buf][ldrow     ][ldcol]);
        async_copy_b128(bptr1 + k0, &Bs[buf][ldrow + 64][ldcol]);
    };

    const int NSTAGES = D_DIM / 16;            // 64
    issue_stage(0, 0);

    for (int s = 0; s < NSTAGES; ++s) {
        wait_async_all();                      // my stage-s data is in LDS
        __syncthreads();                       // everyone's stage-s data visible;
                                               // also orders prev-stage LDS reads
        if (s + 1 < NSTAGES)
            issue_stage((s + 1) & 1, (s + 1) * 16);

        const int cur = s & 1;
        #pragma unroll
        for (int kk = 0; kk < 16; kk += 4) {
            const int krow = kk + 2 * half;
            v2f a;
            a.x = As[cur][16 * wave + idx][krow];
            a.y = As[cur][16 * wave + idx][krow + 1];
            #pragma unroll
            for (int nt = 0; nt < 8; ++nt) {
                v2f b;
                b.x = Bs[cur][16 * nt + idx][krow];
                b.y = Bs[cur][16 * nt + idx][krow + 1];
                acc[nt] = __builtin_amdgcn_wmma_f32_16x16x4_f32(
                    /*neg_a=*/false, a, /*neg_b=*/false, b,
                    /*c_mod=*/(short)0, acc[nt],
                    /*reuse_a=*/false, /*reuse_b=*/false);
            }
        }
    }

    // ---- bias + store; output row stride accounts for (S+1) rows/batch ----
    const int mbase = blockM + 16 * wave + 8 * half;
    #pragma unroll
    for (int nt = 0; nt < 8; ++nt) {
        const int n = blockN + 16 * nt + idx;
        const float bv = bias[n];
        #pragma unroll
        for (int i = 0; i < 8; ++i) {
            const int r = mbase + i;           // global hidden row
            out[(size_t)(r + batch) * D_DIM + n] = acc[nt][i] + bv;
        }
    }
}

// ---------------------------------------------------------------------------
// Kernel 2: scores[b,k] = dot(q_b, key_k) / max(||key_k||, eps); packed
// u64 atomicMax argmax (ties -> smallest key index, matching jnp.argmin).
// One WMMA tile covers all 16 batches x 16 keys; 4 waves/block = 64 keys.
// ---------------------------------------------------------------------------
__global__ __launch_bounds__(128)
void argmin_kernel(const float* __restrict__ q,     // (16, 1024)
                   const float* __restrict__ keys,  // (65536, 1024)
                   unsigned long long* __restrict__ best)
{
    const int tid  = threadIdx.x;
    const int wave = tid >> 5;
    const int lane = tid & 31;
    const int half = lane >> 4;
    const int idx  = lane & 15;
    const int key  = blockIdx.x * 64 + wave * 16 + idx;

    v8f acc = {};
    float nsq = 0.0f;
    const float* qrow = q    + (size_t)idx * D_DIM;
    const float* krow = keys + (size_t)key * D_DIM;

    for (int k0 = 0; k0 < D_DIM; k0 += 4) {
        const int kk = k0 + 2 * half;
        v2f a; a.x = qrow[kk]; a.y = qrow[kk + 1];
        v2f b; b.x = krow[kk]; b.y = krow[kk + 1];
        nsq += b.x * b.x + b.y * b.y;
        acc = __builtin_amdgcn_wmma_f32_16x16x4_f32(
            false, a, false, b, (short)0, acc, false, false);
    }

    // combine the two K-halves of the per-key norm (same key in lane ^ 16)
    nsq += __shfl_xor(nsq, 16);
    const float inv = 1.0f / fmaxf(sqrtf(nsq), 1e-12f);

    #pragma unroll
    for (int i = 0; i < 8; ++i) {
        const int b = i + 8 * half;                 // batch index of this VGPR
        const float score = acc[i] * inv;
        uint32_t fb  = __float_as_uint(score);
        uint32_t ord = (fb & 0x80000000u) ? ~fb : (fb | 0x80000000u);
        unsigned long long packed =
            ((unsigned long long)ord << 32) |
            (unsigned long long)(0xFFFFFFFFu - (uint32_t)key);
        #pragma unroll
        for (int m = 1; m < 16; m <<= 1) {
            unsigned long long other = __shfl_xor(packed, m);
            packed = (packed > other) ? packed : other;
        }
        if (idx == 0) atomicMax(&best[b], packed);
    }
}

// ---------------------------------------------------------------------------
// Kernel 3: tokens = l2norm(values[key_to_value[chosen_key[b]]]), written
// to output row S of batch b. One block per batch.
// ---------------------------------------------------------------------------
__global__ __launch_bounds__(256)
void gather_norm_kernel(const unsigned long long* __restrict__ best,
                        const long long* __restrict__ key_to_value,
                        const float* __restrict__ values,   // (K, 1, D)
                        float* __restrict__ out)
{
    __shared__ float red[8];
    __shared__ float s_scale;
    const int b   = blockIdx.x;
    const int tid = threadIdx.x;

    const uint32_t chosen = 0xFFFFFFFFu - (uint32_t)(best[b] & 0xFFFFFFFFull);
    const long long cidx  = key_to_value[chosen];
    const float* row = values + (size_t)cidx * D_DIM;

    float v[4];
    float s = 0.0f;
    #pragma unroll
    for (int j = 0; j < 4; ++j) {
        v[j] = row[tid + 256 * j];
        s += v[j] * v[j];
    }
    #pragma unroll
    for (int m = 1; m < 32; m <<= 1) s += __shfl_xor(s, m);
    if ((tid & 31) == 0) red[tid >> 5] = s;
    __syncthreads();
    if (tid == 0) {
        float t = 0.0f;
        #pragma unroll
        for (int i = 0; i < 8; ++i) t += red[i];
        s_scale = 1.0f / fmaxf(sqrtf(t), 1e-12f);
    }
    __syncthreads();

    const float sc = s_scale;
    const size_t base = ((size_t)b * (S_DIM + 1) + S_DIM) * D_DIM;
    #pragma unroll
    for (int j = 0; j < 4; ++j) out[base + tid + 256 * j] = v[j] * sc;
}

// ---------------------------------------------------------------------------
extern "C" void kernel_launch(void* const* d_in, const int* in_sizes, int n_in,
                              void* d_out, int out_size, void* d_ws, size_t ws_size,
                              hipStream_t stream) {
    const float*      hidden  = (const float*)d_in[0];       // (B, S, D)
    const float*      concept = (const float*)d_in[1];       // (B, 1, D)
    const float*      keys    = (const float*)d_in[2];       // (K, D)
    const float*      values  = (const float*)d_in[3];       // (K, 1, D)
    const long long*  k2v     = (const long long*)d_in[4];   // (K,) int64
    const float*      W       = (const float*)d_in[5];       // (D, D)
    const float*      bias    = (const float*)d_in[6];       // (D,)
    float*            out     = (float*)d_out;               // (B, S+1, D)
    unsigned long long* best  = (unsigned long long*)d_ws;   // 16 slots

    init_best_kernel<<<1, 16, 0, stream>>>(best);

    dim3 gemm_grid((B_DIM * S_DIM) / 128, D_DIM / 128);      // (512, 8)
    gemm_bias_kernel<<<gemm_grid, 256, 0, stream>>>(hidden, W, bias, out);

    argmin_kernel<<<K_KEYS / 64, 128, 0, stream>>>(concept, keys, best);

    gather_norm_kernel<<<B_DIM, 256, 0, stream>>>(best, k2v, values, out);
}